// ICP_52742198395379
// MI455X (gfx1250) — compile-verified
//
#include <hip/hip_runtime.h>
#include <hip/hip_bf16.h>

#define STEPLIM 8
#define ICP_TOL 1e-4f
#define ICP_EPS 1e-12f

typedef __attribute__((ext_vector_type(2))) float v2f;
typedef __attribute__((ext_vector_type(8))) float v8f;

// ---------------------------------------------------------------------------
// init scan-carry state: err = 0, done = 0, done_apply = 0
__global__ void icp_init(float* errdone) {
    if (threadIdx.x == 0) { errdone[0] = 0.f; errdone[1] = 0.f; errdone[2] = 0.f; }
}

// ---------------------------------------------------------------------------
// 1-NN: one wave handles a 16-row strip of pc for batch b; sweeps M in 32-wide
// chunks (two 16-col tiles) using V_WMMA_F32_16X16X4_F32 for the Gram part
// (K=3 padded to 4). Column-unroll x2 shares the A tile / row norms, issues
// both B-tile b96 loads before one wait, and runs two independent WMMAs for
// in-wave ILP (grid is only ~1 wave/SIMD, so latency hiding must come from ILP).
// A layout (32-bit 16x4): lane c=lane&15 is row M, half h=lane>>4 selects K
// pair {0,1} (x,y) or {2,3} (z,0). B layout mirrors (col N=lane&15).
// C/D layout: vgpr j, lane -> value for row M = 8*h + j, col N = lane&15.
__global__ __launch_bounds__(32) void icp_knn(const float* __restrict__ pc,
                                              const float* __restrict__ p2,
                                              float* __restrict__ vals,
                                              int* __restrict__ idx,
                                              int N, int M) {
    const int tilesPerBatch = N >> 4;
    const int b    = blockIdx.x / tilesPerBatch;
    const int n0   = (blockIdx.x % tilesPerBatch) << 4;
    const int lane = threadIdx.x;
    const int h    = lane >> 4;
    const int c    = lane & 15;

    const float* pr = pc + (size_t)(b * N + n0 + c) * 3;
    float px = pr[0], py = pr[1], pz = pr[2];
    v2f a; a.x = h ? pz : px; a.y = h ? 0.f : py;

    // |pc|^2 for the 8 rows this half of the D tile covers (M = 8*h + j)
    float rn[8];
#pragma unroll
    for (int j = 0; j < 8; ++j) {
        const float* rp = pc + (size_t)(b * N + n0 + 8 * h + j) * 3;
        float x = rp[0], y = rp[1], z = rp[2];
        rn[j] = x * x + y * y + z * z;
    }

    float bestd[8]; int besti[8];
#pragma unroll
    for (int j = 0; j < 8; ++j) { bestd[j] = 3.4e38f; besti[j] = 0; }

    const float* qbase = p2 + (size_t)(b * M + c) * 3;
    for (int m0 = 0; m0 < M; m0 += 32) {
        // two independent 16-col tiles: both loads issue before either is used
        const float* q0 = qbase + (size_t)m0 * 3;
        const float* q1 = q0 + 48;  // +16 points
        float q0x = q0[0], q0y = q0[1], q0z = q0[2];
        float q1x = q1[0], q1y = q1[1], q1z = q1[2];
        float qn0 = q0x * q0x + q0y * q0y + q0z * q0z;
        float qn1 = q1x * q1x + q1y * q1y + q1z * q1z;
        v2f bv0; bv0.x = h ? q0z : q0x; bv0.y = h ? 0.f : q0y;
        v2f bv1; bv1.x = h ? q1z : q1x; bv1.y = h ? 0.f : q1y;
        v8f acc0 = {};
        v8f acc1 = {};
        acc0 = __builtin_amdgcn_wmma_f32_16x16x4_f32(false, a, false, bv0,
                                                     (short)0, acc0, false, false);
        acc1 = __builtin_amdgcn_wmma_f32_16x16x4_f32(false, a, false, bv1,
                                                     (short)0, acc1, false, false);
        int mi0 = m0 + c;
        int mi1 = m0 + 16 + c;
#pragma unroll
        for (int j = 0; j < 8; ++j) {
            float d2a = rn[j] + qn0 - 2.f * acc0[j];   // monotone in d=sqrt(...)
            float d2b = rn[j] + qn1 - 2.f * acc1[j];
            if (d2a < bestd[j]) { bestd[j] = d2a; besti[j] = mi0; }
            if (d2b < bestd[j]) { bestd[j] = d2b; besti[j] = mi1; }
        }
    }

    // argmin across the 16 lanes of each half (xor masks stay inside the half)
#pragma unroll
    for (int off = 8; off > 0; off >>= 1) {
#pragma unroll
        for (int j = 0; j < 8; ++j) {
            float od = __shfl_xor(bestd[j], off, 32);
            int   oi = __shfl_xor(besti[j], off, 32);
            if (od < bestd[j] || (od == bestd[j] && oi < besti[j])) {
                bestd[j] = od; besti[j] = oi;
            }
        }
    }

    if (c == 0) {
#pragma unroll
        for (int j = 0; j < 8; ++j) {
            int row = n0 + 8 * h + j;
            vals[b * N + row] = sqrtf(fmaxf(bestd[j], ICP_EPS));
            idx[b * N + row]  = besti[j];
        }
    }
}

// ---------------------------------------------------------------------------
// per-batch sums: [0]=sum vals, [1..3]=sum src, [4..6]=sum tgt, [7..15]=sum src_i*tgt_j
// tgt gathered with LAST batch's indices (reference uses idx[-1]) when gidx!=null.
__global__ void icp_stats(const float* __restrict__ src,
                          const float* __restrict__ tgt,
                          const int* __restrict__ gidx,
                          const float* __restrict__ vals,
                          float* __restrict__ partial,
                          int N, int Mt, int B) {
    const int b = blockIdx.x;
    const int tid = threadIdx.x;
    float acc[16];
#pragma unroll
    for (int k = 0; k < 16; ++k) acc[k] = 0.f;
    for (int n = tid; n < N; n += 256) {
        const float* s = src + (size_t)(b * N + n) * 3;
        int tn = gidx ? gidx[(size_t)(B - 1) * N + n] : n;
        const float* t = tgt + (size_t)(b * Mt + tn) * 3;
        float sx = s[0], sy = s[1], sz = s[2];
        float tx = t[0], ty = t[1], tz = t[2];
        if (vals) acc[0] += vals[b * N + n];
        acc[1] += sx; acc[2] += sy; acc[3] += sz;
        acc[4] += tx; acc[5] += ty; acc[6] += tz;
        acc[7]  += sx * tx; acc[8]  += sx * ty; acc[9]  += sx * tz;
        acc[10] += sy * tx; acc[11] += sy * ty; acc[12] += sy * tz;
        acc[13] += sz * tx; acc[14] += sz * ty; acc[15] += sz * tz;
    }
    __shared__ float sh[256 * 16];
#pragma unroll
    for (int k = 0; k < 16; ++k) sh[tid * 16 + k] = acc[k];
    __syncthreads();
    for (int s = 128; s > 0; s >>= 1) {
        if (tid < s) {
#pragma unroll
            for (int k = 0; k < 16; ++k) sh[tid * 16 + k] += sh[(tid + s) * 16 + k];
        }
        __syncthreads();
    }
    if (tid == 0) {
#pragma unroll
        for (int k = 0; k < 16; ++k) partial[b * 16 + k] = sh[k];
    }
}

// ---------------------------------------------------------------------------
__device__ inline float det3(const float M[3][3]) {
    return M[0][0] * (M[1][1] * M[2][2] - M[1][2] * M[2][1])
         - M[0][1] * (M[1][0] * M[2][2] - M[1][2] * M[2][0])
         + M[0][2] * (M[1][0] * M[2][1] - M[1][1] * M[2][0]);
}

// Kabsch rotation from 3x3 correlation H: H = U S V^T, R = V diag(1,1,det) U^T.
// SVD via cyclic Jacobi on H^T H.
__device__ void kabsch_rotation(const float H[3][3], float R[3][3]) {
    float Bm[3][3];
    for (int i = 0; i < 3; ++i)
        for (int j = 0; j < 3; ++j)
            Bm[i][j] = H[0][i] * H[0][j] + H[1][i] * H[1][j] + H[2][i] * H[2][j];
    float V[3][3] = {{1, 0, 0}, {0, 1, 0}, {0, 0, 1}};
    const int pp[3] = {0, 0, 1}, qq[3] = {1, 2, 2};
    for (int sweep = 0; sweep < 15; ++sweep) {
        for (int r = 0; r < 3; ++r) {
            int p = pp[r], q = qq[r];
            float apq = Bm[p][q];
            if (fabsf(apq) > 1e-20f) {
                float tau = (Bm[q][q] - Bm[p][p]) / (2.f * apq);
                float tt = (tau >= 0.f ? 1.f : -1.f) / (fabsf(tau) + sqrtf(1.f + tau * tau));
                float cc = 1.f / sqrtf(1.f + tt * tt);
                float ss = tt * cc;
                for (int k = 0; k < 3; ++k) {
                    float bkp = Bm[k][p], bkq = Bm[k][q];
                    Bm[k][p] = cc * bkp - ss * bkq;
                    Bm[k][q] = ss * bkp + cc * bkq;
                }
                for (int k = 0; k < 3; ++k) {
                    float bpk = Bm[p][k], bqk = Bm[q][k];
                    Bm[p][k] = cc * bpk - ss * bqk;
                    Bm[q][k] = ss * bpk + cc * bqk;
                }
                for (int k = 0; k < 3; ++k) {
                    float vkp = V[k][p], vkq = V[k][q];
                    V[k][p] = cc * vkp - ss * vkq;
                    V[k][q] = ss * vkp + cc * vkq;
                }
            }
        }
    }
    float lam[3] = {Bm[0][0], Bm[1][1], Bm[2][2]};
    int o0 = 0, o1 = 1, o2 = 2, tswp;
    if (lam[o0] < lam[o1]) { tswp = o0; o0 = o1; o1 = tswp; }
    if (lam[o1] < lam[o2]) { tswp = o1; o1 = o2; o2 = tswp; }
    if (lam[o0] < lam[o1]) { tswp = o0; o0 = o1; o1 = tswp; }
    int ord[3] = {o0, o1, o2};
    float Vs[3][3], U[3][3], sv[3];
    for (int k = 0; k < 3; ++k) {
        for (int i = 0; i < 3; ++i) Vs[i][k] = V[i][ord[k]];
        sv[k] = sqrtf(fmaxf(lam[ord[k]], 0.f));
    }
    const float tinyS = 1e-12f;
    if (sv[0] <= tinyS) {  // H ~ 0
        for (int i = 0; i < 3; ++i)
            for (int j = 0; j < 3; ++j) R[i][j] = (i == j) ? 1.f : 0.f;
        return;
    }
    for (int k = 0; k < 2; ++k) {
        if (sv[k] > tinyS) {
            float inv = 1.f / sv[k];
            for (int i = 0; i < 3; ++i)
                U[i][k] = (H[i][0] * Vs[0][k] + H[i][1] * Vs[1][k] + H[i][2] * Vs[2][k]) * inv;
        } else {  // k==1 degenerate: any unit vector orthogonal to U col 0
            float ax = fabsf(U[0][0]);
            float e0 = (ax < 0.9f) ? 1.f : 0.f, e1 = 1.f - e0;
            float cx = e1 * U[2][0];
            float cy = -e0 * U[2][0];
            float cz = e0 * U[1][0] - e1 * U[0][0];
            float il = 1.f / sqrtf(fmaxf(cx * cx + cy * cy + cz * cz, 1e-30f));
            U[0][1] = cx * il; U[1][1] = cy * il; U[2][1] = cz * il;
        }
    }
    if (sv[2] > tinyS) {
        float inv = 1.f / sv[2];
        for (int i = 0; i < 3; ++i)
            U[i][2] = (H[i][0] * Vs[0][2] + H[i][1] * Vs[1][2] + H[i][2] * Vs[2][2]) * inv;
    } else {
        float cx = U[1][0] * U[2][1] - U[2][0] * U[1][1];
        float cy = U[2][0] * U[0][1] - U[0][0] * U[2][1];
        float cz = U[0][0] * U[1][1] - U[1][0] * U[0][1];
        float il = 1.f / sqrtf(fmaxf(cx * cx + cy * cy + cz * cz, 1e-30f));
        U[0][2] = cx * il; U[1][2] = cy * il; U[2][2] = cz * il;
    }
    float dsign = det3(Vs) * det3(U);
    float D[3] = {1.f, 1.f, dsign};
    for (int i = 0; i < 3; ++i)
        for (int j = 0; j < 3; ++j)
            R[i][j] = Vs[i][0] * D[0] * U[j][0] + Vs[i][1] * D[1] * U[j][1]
                    + Vs[i][2] * D[2] * U[j][2];
}

// mode 0: scan-step update (err/done gating + per-batch R,t into ws)
// mode 1: final transform, write [R|t] (B x 3 x 4) to out
__global__ void icp_solve(const float* __restrict__ partial,
                          float* __restrict__ Rt,
                          float* __restrict__ errdone,
                          float* __restrict__ out,
                          int N, int B, int mode) {
    int tid = threadIdx.x;
    if (mode == 0 && tid == 0) {
        float sum = 0.f;
        for (int b = 0; b < B; ++b) sum += partial[b * 16 + 0];
        float errnew  = sum / (float)B;         // sum(mean(vals, axis=0))
        float errold  = errdone[0];
        float doneold = errdone[1];
        errdone[2] = doneold;                   // gate used by apply this step
        float donenew = (doneold != 0.f || fabsf(errold - errnew) < ICP_TOL) ? 1.f : 0.f;
        errdone[0] = (doneold != 0.f) ? errold : errnew;
        errdone[1] = donenew;
    }
    if (tid < B) {
        int b = tid;
        const float* P = partial + b * 16;
        float invN = 1.f / (float)N;
        float c1[3] = {P[1] * invN, P[2] * invN, P[3] * invN};
        float c2[3] = {P[4] * invN, P[5] * invN, P[6] * invN};
        float H[3][3];
        for (int i = 0; i < 3; ++i)
            for (int j = 0; j < 3; ++j)
                H[i][j] = P[7 + i * 3 + j] - (float)N * c1[i] * c2[j];
        float R[3][3];
        kabsch_rotation(H, R);
        float t[3];
        for (int i = 0; i < 3; ++i)
            t[i] = c2[i] - (R[i][0] * c1[0] + R[i][1] * c1[1] + R[i][2] * c1[2]);
        if (mode == 0) {
            float* Q = Rt + b * 12;
            for (int i = 0; i < 3; ++i)
                for (int j = 0; j < 3; ++j) Q[i * 3 + j] = R[i][j];
            Q[9] = t[0]; Q[10] = t[1]; Q[11] = t[2];
        } else {
            for (int i = 0; i < 3; ++i) {
                for (int j = 0; j < 3; ++j) out[b * 12 + i * 4 + j] = R[i][j];
                out[b * 12 + i * 4 + 3] = t[i];
            }
        }
    }
}

// ---------------------------------------------------------------------------
// pc <- done ? pc : R*pc + t   (gate uses done from BEFORE this step's update)
__global__ void icp_apply(float* __restrict__ pc, const float* __restrict__ Rt,
                          const float* __restrict__ errdone, int N, int B) {
    if (errdone[2] != 0.f) return;
    int gid = blockIdx.x * blockDim.x + threadIdx.x;
    if (gid >= B * N) return;
    int b = gid / N;
    const float* Q = Rt + b * 12;
    float* p = pc + (size_t)gid * 3;
    float x = p[0], y = p[1], z = p[2];
    float nx = Q[0] * x + Q[1] * y + Q[2] * z + Q[9];
    float ny = Q[3] * x + Q[4] * y + Q[5] * z + Q[10];
    float nz = Q[6] * x + Q[7] * y + Q[8] * z + Q[11];
    p[0] = nx; p[1] = ny; p[2] = nz;
}

// ---------------------------------------------------------------------------
extern "C" void kernel_launch(void* const* d_in, const int* in_sizes, int n_in,
                              void* d_out, int out_size, void* d_ws, size_t ws_size,
                              hipStream_t stream) {
    (void)n_in; (void)out_size; (void)ws_size;
    const float* p1 = (const float*)d_in[0];
    const float* p2 = (const float*)d_in[1];
    const int B = 2;
    const int N = in_sizes[0] / (B * 3);
    const int M = in_sizes[1] / (B * 3);

    // workspace layout (floats / ints, 4-byte aligned)
    float* ws      = (float*)d_ws;
    float* pc      = ws;                               // B*N*3
    float* vals    = pc + (size_t)B * N * 3;           // B*N
    int*   idx     = (int*)(vals + (size_t)B * N);     // B*N
    float* partial = (float*)(idx + (size_t)B * N);    // B*16
    float* Rt      = partial + B * 16;                 // B*12
    float* errdone = Rt + B * 12;                      // 3

    hipMemcpyAsync(pc, p1, sizeof(float) * (size_t)B * N * 3,
                   hipMemcpyDeviceToDevice, stream);
    icp_init<<<1, 32, 0, stream>>>(errdone);

    const int knnGrid = B * (N / 16);
    for (int it = 0; it <= STEPLIM; ++it) {
        icp_knn<<<knnGrid, 32, 0, stream>>>(pc, p2, vals, idx, N, M);
        icp_stats<<<B, 256, 0, stream>>>(pc, p2, idx, vals, partial, N, M, B);
        icp_solve<<<1, 32, 0, stream>>>(partial, Rt, errdone, nullptr, N, B, 0);
        icp_apply<<<(B * N + 255) / 256, 256, 0, stream>>>(pc, Rt, errdone, N, B);
    }
    // final: R,t = points_transform(p1, pc_final) -> [R|t] out (2x3x4)
    icp_stats<<<B, 256, 0, stream>>>(p1, pc, nullptr, nullptr, partial, N, N, B);
    icp_solve<<<1, 32, 0, stream>>>(partial, Rt, errdone, (float*)d_out, N, B, 1);
}